// CapsuleLayer_10814727651666
// MI455X (gfx1250) — compile-verified
//
#include <hip/hip_runtime.h>
#include <math.h>

typedef __attribute__((ext_vector_type(2))) float v2f;
typedef __attribute__((ext_vector_type(4))) float v4f;
typedef __attribute__((ext_vector_type(8))) float v8f;

// Problem constants (from reference)
constexpr int kB   = 32;    // batch (GEMM N)
constexpr int kIU  = 16;    // in_units
constexpr int kIC  = 2048;  // in_channels
constexpr int kNU  = 16;    // num_units
constexpr int kUS  = 64;    // unit_size
constexpr int kKS  = 8;     // K-splits across workgroups
// GEMM view: M = kNU*kUS = 1024, K = kIC*kIU = 32768, N = kB = 32
// Wmat[m=(j*64+k)][kk=(i*16+u)] = W[i,j,k,u]  (addr = i*16384 + j*1024 + k*16 + u)
//
// K windows per i (4 WMMAs): window s covers u in {2s, 2s+1, 2s+8, 2s+9}
//   WMMA K0,K1 (lanes 0-15)  <- u = 2s, 2s+1       (kgrp=0)
//   WMMA K2,K3 (lanes 16-31) <- u = 2s+8, 2s+9     (kgrp=1)
// => lane's A data for windows {0,1} is the contiguous float4 at u = 8*kgrp,
//    and for windows {2,3} at u = 8*kgrp + 4.
//
// Xt2 layout (B fragments, register order):
//   float index = ((i*2 + tile)*2 + half)*128 + lane*4 + slo*2 + r
//   where s = half*2 + slo, lane gives (kgrp, n=lane&15 + 16*tile),
//   value = x[n, u=2s+8*kgrp+r, i]
//
// Workspace: [0,1MB) partials[kKS][1024][32] ; [1MB,5MB) Xt2 (1M floats)
constexpr size_t kPartialsElems = (size_t)kKS * 1024 * kB;   // 262144 floats

// ---------------- Kernel A: pack x -> Xt2 (B-fragment order, coalesced writes) ----
__global__ __launch_bounds__(256) void caps_xt(const float* __restrict__ x,
                                               float* __restrict__ xt2) {
  int idx  = blockIdx.x * 256 + threadIdx.x;   // 0 .. 2048*512-1
  int r    = idx & 1;
  int slo  = (idx >> 1) & 1;
  int lane = (idx >> 2) & 31;
  int half = (idx >> 7) & 1;
  int tile = (idx >> 8) & 1;
  int i    = idx >> 9;
  int s    = half * 2 + slo;
  int kgrp = lane >> 4;
  int n    = (lane & 15) + 16 * tile;
  int u    = 2 * s + 8 * kgrp + r;
  xt2[idx] = x[(size_t)n * (kIU * kIC) + (size_t)u * kIC + i];
}

// ---------------- Kernel B: WMMA GEMM, K split over blockIdx.y --------------------
__global__ __launch_bounds__(256) void caps_gemm(const float* __restrict__ xt2,
                                                 const float* __restrict__ W,
                                                 float* __restrict__ partials) {
  const int lane  = threadIdx.x & 31;
  const int wave  = threadIdx.x >> 5;   // 0..7
  const int mtile = blockIdx.x;         // 0..63 : 16 rows of M
  const int ksp   = blockIdx.y;         // 0..7  : i-range [ksp*256, ksp*256+256)
  const int j     = mtile >> 2;
  const int k0    = (mtile & 3) << 4;
  const int mrow  = lane & 15;
  const int kgrp  = lane >> 4;

  v8f acc0 = {};  // b = 0..15
  v8f acc1 = {};  // b = 16..31

  const float* Wjk = W + (size_t)j * (kUS * kIU) + (size_t)k0 * kIU
                       + (size_t)mrow * kIU + (size_t)kgrp * 8;
  const int i0 = ksp * (kIC / kKS);

  for (int t = 0; t < (kIC / kKS) / 8; ++t) {   // 32 iterations per wave
    const int i = i0 + wave + t * 8;
    const float* Wb = Wjk + (size_t)i * (kNU * kUS * kIU);
    __builtin_prefetch(Wb + (size_t)8 * (kNU * kUS * kIU), 0, 1);
    // A fragments: windows {0,1} then {2,3}; streamed once -> non-temporal
    v4f a01 = __builtin_nontemporal_load((const v4f*)(Wb));
    v4f a23 = __builtin_nontemporal_load((const v4f*)(Wb + 4));
    // B fragments: dense 512B blocks per (tile, half)
    const float* Xb = xt2 + (size_t)i * 512 + lane * 4;
    v4f b0a = *(const v4f*)(Xb);         // tile0, s=0,1
    v4f b0b = *(const v4f*)(Xb + 128);   // tile0, s=2,3
    v4f b1a = *(const v4f*)(Xb + 256);   // tile1, s=0,1
    v4f b1b = *(const v4f*)(Xb + 384);   // tile1, s=2,3

    v2f a, b;
#define STEP(AV, LOHI, B0, B1)                                             \
    a = __builtin_shufflevector(AV, AV, 0 + LOHI, 1 + LOHI);               \
    b = __builtin_shufflevector(B0, B0, 0 + LOHI, 1 + LOHI);               \
    acc0 = __builtin_amdgcn_wmma_f32_16x16x4_f32(false, a, false, b,       \
                                      (short)0, acc0, false, false);       \
    b = __builtin_shufflevector(B1, B1, 0 + LOHI, 1 + LOHI);               \
    acc1 = __builtin_amdgcn_wmma_f32_16x16x4_f32(false, a, false, b,       \
                                      (short)0, acc1, false, false);
    STEP(a01, 0, b0a, b1a)   // window s=0
    STEP(a01, 2, b0a, b1a)   // window s=1
    STEP(a23, 0, b0b, b1b)   // window s=2
    STEP(a23, 2, b0b, b1b)   // window s=3
#undef STEP
  }

  // Cross-wave reduction in LDS: 8 waves x 2 tiles x 256 floats = 16 KB
  __shared__ float lds[8][2][256];
  *(v8f*)&lds[wave][0][lane * 8] = acc0;
  *(v8f*)&lds[wave][1][lane * 8] = acc1;
  __syncthreads();

  int flat = threadIdx.x * 2;
#pragma unroll
  for (int q = 0; q < 2; ++q, ++flat) {
    const int t = flat >> 8;        // N-tile
    const int p = flat & 255;       // (lane, vgpr) slot
    float sum = 0.f;
#pragma unroll
    for (int w = 0; w < 8; ++w) sum += lds[w][t][p];
    // C/D layout: lane l, vgpr r -> row = r + 8*(l>>4), col = l&15
    const int l  = p >> 3, r = p & 7;
    const int mr = r + ((l >> 4) << 3);
    const int n  = t * 16 + (l & 15);
    partials[(size_t)ksp * (1024 * kB) + (size_t)(mtile * 16 + mr) * kB + n] = sum;
  }
}

// ---------------- Kernel C: reduce K-split partials + squash ----------------------
__global__ __launch_bounds__(64) void caps_squash(const float* __restrict__ partials,
                                                  float* __restrict__ out) {
  const int b = blockIdx.x;   // 0..31
  const int k = threadIdx.x;  // 0..63
  float sv[kNU];
  float mag_sq = 0.f;
#pragma unroll
  for (int j = 0; j < kNU; ++j) {
    float v = 0.f;
#pragma unroll
    for (int p = 0; p < kKS; ++p)
      v += partials[(size_t)p * (1024 * kB) + (size_t)(j * kUS + k) * kB + b];
    sv[j] = v;
    mag_sq += v * v;
  }
  const float mag   = sqrtf(mag_sq);
  const float scale = mag_sq / ((1.f + mag_sq) * mag);
#pragma unroll
  for (int j = 0; j < kNU; ++j)
    out[(size_t)b * (kNU * kUS) + j * kUS + k] = sv[j] * scale;
}

extern "C" void kernel_launch(void* const* d_in, const int* in_sizes, int n_in,
                              void* d_out, int out_size, void* d_ws, size_t ws_size,
                              hipStream_t stream) {
  const float* x = (const float*)d_in[0];  // (32, 16, 2048)
  const float* W = (const float*)d_in[1];  // (1, 2048, 16, 64, 16)
  float* out      = (float*)d_out;         // (32, 16, 64)
  float* partials = (float*)d_ws;                   // 1 MB
  float* xt2      = (float*)d_ws + kPartialsElems;  // 4 MB

  caps_xt<<<(2048 * 512) / 256, 256, 0, stream>>>(x, xt2);
  dim3 grid(64, kKS);
  caps_gemm<<<grid, 256, 0, stream>>>(xt2, W, partials);
  caps_squash<<<32, 64, 0, stream>>>(partials, out);
}